// SymmetricNormLinkPredictor_67190468378858
// MI455X (gfx1250) — compile-verified
//
#include <hip/hip_runtime.h>

typedef __attribute__((ext_vector_type(2))) float v2f;
typedef __attribute__((ext_vector_type(8))) float v8f;

#define N_NODES 50000
#define N_EDGES 800000
#define N_QUERY 200000

// ---------------- degree / normalization ----------------

__global__ void deg_init_kernel(float* deg, int n) {
  int i = blockIdx.x * blockDim.x + threadIdx.x;
  if (i < n) deg[i] = 1.0f;  // self loop contributes 1 to every node's row-count
}

__global__ void deg_count_kernel(const int* __restrict__ row, float* deg, int e) {
  int i = blockIdx.x * blockDim.x + threadIdx.x;
  if (i < e) atomicAdd(&deg[row[i]], 1.0f);
}

__global__ void deg_rsqrt_kernel(float* deg, int n) {
  int i = blockIdx.x * blockDim.x + threadIdx.x;
  if (i < n) deg[i] = rsqrtf(deg[i]);  // in-place: deg -> dis
}

// ---------------- GCN scatter (message passing) ----------------

// Initialize agg with the self-loop message: agg[i] = x[i] * dis[i]^2
template<int D>
__global__ void selfloop_init_kernel(const float* __restrict__ x, const float* __restrict__ dis,
                                     float* __restrict__ agg, int n) {
  const int CH = D / 4;
  int t = blockIdx.x * blockDim.x + threadIdx.x;
  int node = t / CH;
  int k = (t % CH) * 4;
  if (node >= n) return;
  float s = dis[node];
  s = s * s;
  const float4 v = *(const float4*)(x + (size_t)node * D + k);
  float4 o;
  o.x = v.x * s; o.y = v.y * s; o.z = v.z * s; o.w = v.w * s;
  *(float4*)(agg + (size_t)node * D + k) = o;
}

// agg[col[e]] += feat[row[e]] * dis[row]*dis[col]   (4 floats per thread)
template<int D>
__global__ void edge_scatter_kernel(const float* __restrict__ feat, const int* __restrict__ row,
                                    const int* __restrict__ col, const float* __restrict__ dis,
                                    float* __restrict__ agg, int e) {
  const int CH = D / 4;
  int t = blockIdx.x * blockDim.x + threadIdx.x;
  int edge = t / CH;
  int k = (t % CH) * 4;
  if (edge >= e) return;
  int r = row[edge], c = col[edge];
  float norm = dis[r] * dis[c];
  const float4 v = *(const float4*)(feat + (size_t)r * D + k);
  float* dst = agg + (size_t)c * D + k;
  atomicAdd(dst + 0, v.x * norm);
  atomicAdd(dst + 1, v.y * norm);
  atomicAdd(dst + 2, v.z * norm);
  atomicAdd(dst + 3, v.w * norm);
}

// ---------------- WMMA f32 GEMM: out[N,128] = A[N,K] @ W[K,128] + bias ----------------
// Block = 256 threads = 8 waves. Block owns 16 rows; wave w owns columns [w*16, w*16+16).
// V_WMMA_F32_16X16X4_F32 VGPR layout (ISA 7.12.2):
//   A frag (2 VGPRs): lane half h = lane>>4, m = lane&15; a[j] = A[m][k0 + j + 2h]
//   B frag (2 VGPRs): b[j] = W[k0 + j + 2h][n],  n = lane&15
//   C/D  (8 VGPRs):  c[v] = C[v + 8h][n]
template<int K, bool RELU>
__global__ void gemm_node_kernel(const float* __restrict__ A, const float* __restrict__ W,
                                 const float* __restrict__ bias, float* __restrict__ out) {
  const int wave = threadIdx.x >> 5;
  const int lane = threadIdx.x & 31;
  const int half = lane >> 4;
  const int mn = lane & 15;
  const int row0 = blockIdx.x * 16;
  const int col = wave * 16 + mn;
  const float* Arow = A + (size_t)(row0 + mn) * K;
  v8f acc = {};
  for (int k0 = 0; k0 < K; k0 += 4) {
    const int ka = k0 + half * 2;
    v2f a;
    a.x = Arow[ka];
    a.y = Arow[ka + 1];
    v2f b;
    b.x = W[(size_t)ka * 128 + col];
    b.y = W[(size_t)(ka + 1) * 128 + col];
    acc = __builtin_amdgcn_wmma_f32_16x16x4_f32(false, a, false, b, (short)0, acc,
                                                false, false);
  }
  const float bb = bias[col];
#pragma unroll
  for (int v = 0; v < 8; v++) {
    float r = acc[v] + bb;
    if (RELU) r = fmaxf(r, 0.0f);
    out[(size_t)(row0 + v + half * 8) * 128 + col] = r;
  }
}

// ---------------- fused link predictor ----------------
// Block = 16 queries. A row m = concat(h2[src[q0+m]], h2[dst[q0+m]]) (K=256).
// Wave w computes hidden cols [w*16,+16) via WMMA, applies bias+ReLU, dots with Wc2,
// reduces across the 16 lanes of each half (shfl_xor stays within half for masks<16),
// then combines the 8 wave partials deterministically through LDS.
__global__ void predict_kernel(const float* __restrict__ h2, const int* __restrict__ eli,
                               const float* __restrict__ Wc1, const float* __restrict__ bc1,
                               const float* __restrict__ Wc2, const float* __restrict__ bc2,
                               float* __restrict__ out) {
  __shared__ float red[8][16];
  const int wave = threadIdx.x >> 5;
  const int lane = threadIdx.x & 31;
  const int half = lane >> 4;
  const int mn = lane & 15;
  const int q0 = blockIdx.x * 16;
  const int q = q0 + mn;
  const int s = eli[q];
  const int d = eli[N_QUERY + q];
  const float* rs = h2 + (size_t)s * 128;
  const float* rd = h2 + (size_t)d * 128;
  const int col = wave * 16 + mn;
  v8f acc = {};
  for (int k0 = 0; k0 < 256; k0 += 4) {
    const int ka = k0 + half * 2;
    v2f a;
    a.x = (ka < 128) ? rs[ka] : rd[ka - 128];
    a.y = (ka + 1 < 128) ? rs[ka + 1] : rd[ka + 1 - 128];
    v2f b;
    b.x = Wc1[(size_t)ka * 128 + col];
    b.y = Wc1[(size_t)(ka + 1) * 128 + col];
    acc = __builtin_amdgcn_wmma_f32_16x16x4_f32(false, a, false, b, (short)0, acc,
                                                false, false);
  }
  const float bb = bc1[col];
  const float w2 = Wc2[col];
#pragma unroll
  for (int v = 0; v < 8; v++) {
    float p = fmaxf(acc[v] + bb, 0.0f) * w2;   // ReLU + final-layer weight
    p += __shfl_xor(p, 1, 32);
    p += __shfl_xor(p, 2, 32);
    p += __shfl_xor(p, 4, 32);
    p += __shfl_xor(p, 8, 32);                 // sum over 16 cols within this half
    if (mn == 0) red[wave][v + half * 8] = p;  // partial for query row v+8*half
  }
  __syncthreads();
  if (threadIdx.x < 16) {
    float sum = bc2[0];
#pragma unroll
    for (int w = 0; w < 8; w++) sum += red[w][threadIdx.x];
    out[q0 + threadIdx.x] = sum;
  }
}

// ---------------- launcher ----------------

extern "C" void kernel_launch(void* const* d_in, const int* in_sizes, int n_in,
                              void* d_out, int out_size, void* d_ws, size_t ws_size,
                              hipStream_t stream) {
  (void)in_sizes; (void)n_in; (void)out_size; (void)ws_size;
  const float* x   = (const float*)d_in[0];
  const int*   ei  = (const int*)d_in[1];    // [2, E]
  const int*   eli = (const int*)d_in[2];    // [2, Q]
  const float* W1  = (const float*)d_in[3];  // [64,128]
  const float* b1  = (const float*)d_in[4];
  const float* W2  = (const float*)d_in[5];  // [128,128]
  const float* b2  = (const float*)d_in[6];
  const float* Wc1 = (const float*)d_in[7];  // [256,128]
  const float* bc1 = (const float*)d_in[8];
  const float* Wc2 = (const float*)d_in[9];  // [128,1]
  const float* bc2 = (const float*)d_in[10];
  float* out = (float*)d_out;

  float* ws   = (float*)d_ws;
  float* dis  = ws;                    // N floats (deg -> dis in place)
  float* bufA = ws + 50048;            // 128*N floats: agg1, later h2
  float* bufB = bufA + 6400000;        // 128*N floats: h1
  float* bufC = bufB + 6400000;        // 128*N floats: agg2

  const int* row = ei;
  const int* col = ei + N_EDGES;
  const int T = 256;

  // degree + normalization
  deg_init_kernel<<<(N_NODES + T - 1) / T, T, 0, stream>>>(dis, N_NODES);
  deg_count_kernel<<<(N_EDGES + T - 1) / T, T, 0, stream>>>(row, dis, N_EDGES);
  deg_rsqrt_kernel<<<(N_NODES + T - 1) / T, T, 0, stream>>>(dis, N_NODES);

  // conv1: scatter at 64-wide, then GEMM 50000x64x128 + ReLU
  selfloop_init_kernel<64><<<(N_NODES * 16) / T, T, 0, stream>>>(x, dis, bufA, N_NODES);
  edge_scatter_kernel<64><<<(N_EDGES * 16) / T, T, 0, stream>>>(x, row, col, dis, bufA, N_EDGES);
  gemm_node_kernel<64, true><<<N_NODES / 16, T, 0, stream>>>(bufA, W1, b1, bufB);

  // conv2: scatter at 128-wide, then GEMM 50000x128x128 (no ReLU); h2 -> bufA
  selfloop_init_kernel<128><<<(N_NODES * 32) / T, T, 0, stream>>>(bufB, dis, bufC, N_NODES);
  edge_scatter_kernel<128><<<(N_EDGES * 32) / T, T, 0, stream>>>(bufB, row, col, dis, bufC, N_EDGES);
  gemm_node_kernel<128, false><<<N_NODES / 16, T, 0, stream>>>(bufC, W2, b2, bufA);

  // fused predictor: concat-gather + GEMM(Q x 256 x 128) + ReLU + 128->1 dot
  predict_kernel<<<N_QUERY / 16, T, 0, stream>>>(bufA, eli, Wc1, bc1, Wc2, bc2, out);
}